// NCEAverage_7722351198724
// MI455X (gfx1250) — compile-verified
//
#include <hip/hip_runtime.h>
#include <math.h>

#define BATCH 128
#define DIM 128
#define KP1 4097          // K + 1
#define NTILE 257         // ceil(4097 / 16)
#define WAVES_PER_BLOCK 4
#define NROWS 200000
#define MEMSZ ((long long)NROWS * DIM)
#define OUTS_SZ (6LL * BATCH * KP1)
#define PAD 4             // LDS row padding (floats) -> conflict-free b64 A reads

typedef __attribute__((ext_vector_type(2))) float v2f;
typedef __attribute__((ext_vector_type(8))) float v8f;

// ---------------------------------------------------------------------------
// Score kernel: D(16x16) = A(16x4) x B(4x16) + C, accumulated 32x over K=128.
// A = 16 gathered memory rows (one k-tile); B col 0 = f1, col 1 = f2, other
// columns read a zeroed LDS row (branch-free). Pure f32 WMMA.
// ---------------------------------------------------------------------------
__global__ __launch_bounds__(WAVES_PER_BLOCK * 32) void score_kernel(
    const float* __restrict__ l, const float* __restrict__ ab,
    const float* __restrict__ ori, const int* __restrict__ idx,
    const float* __restrict__ mem_l, const float* __restrict__ mem_ab,
    const float* __restrict__ mem_ori, float* __restrict__ outs)
{
    const int wave = threadIdx.x >> 5;
    const int lane = threadIdx.x & 31;
    const int b    = blockIdx.y;
    const int bank = blockIdx.z;

    const float* mem; const float* f1; const float* f2; int slot0, slot1;
    if (bank == 0)      { mem = mem_l;   f1 = ab; f2 = ori; slot0 = 0; slot1 = 2; } // ab2l, ori2l
    else if (bank == 1) { mem = mem_ab;  f1 = l;  f2 = ori; slot0 = 1; slot1 = 5; } // l2ab, ori2ab
    else                { mem = mem_ori; f1 = l;  f2 = ab;  slot0 = 3; slot1 = 4; } // l2ori, ab2ori

    __shared__ __align__(16) float sF1[DIM];
    __shared__ __align__(16) float sF2[DIM];
    __shared__ __align__(16) float sZero[DIM];
    __shared__ __align__(16) float sRows[WAVES_PER_BLOCK][16][DIM + PAD];

    // blockDim.x == DIM == 128: every thread loads one element, no guard
    sF1[threadIdx.x]   = f1[b * DIM + threadIdx.x];
    sF2[threadIdx.x]   = f2[b * DIM + threadIdx.x];
    sZero[threadIdx.x] = 0.0f;

    const int kt = blockIdx.x * WAVES_PER_BLOCK + wave;   // wave-uniform
    if (kt < NTILE) {
        // lanes 0..15 fetch the gather indices for the 16 rows of this tile
        int rowidx = 0;
        if (lane < 16) {
            int k  = kt * 16 + lane;
            int kk = (k < KP1) ? k : (KP1 - 1);
            rowidx = idx[b * KP1 + kk];
        }
        // coalesced gather: one 512B row per pass (lane loads float4)
        for (int r = 0; r < 16; ++r) {
            int ri = __shfl(rowidx, r, 32);
            float4 v = *(const float4*)(mem + (long long)ri * DIM + lane * 4);
            *(float4*)(&sRows[wave][r][lane * 4]) = v;
        }
    }
    __syncthreads();

    if (kt < NTILE) {
        v8f acc = {};
        const int arow = lane & 15;        // M index (A rows) / N index (B cols)
        const int chi  = lane >> 4;        // 0: K pair {0,1}, 1: K pair {2,3}
        const int n    = lane & 15;

        // Loop-invariant, branch-free fragment base pointers (selected once).
        const float* __restrict__ aptr = &sRows[wave][arow][2 * chi];
        const float* __restrict__ bptr =
            ((n == 0) ? sF1 : ((n == 1) ? sF2 : sZero)) + 2 * chi;

        // Software-pipelined: fetch s+1 fragments before issuing WMMA for s.
        v2f a  = *(const v2f*)(aptr);
        v2f bf = *(const v2f*)(bptr);
        #pragma unroll
        for (int s = 0; s < 32; ++s) {
            v2f an, bn;
            if (s < 31) {
                an = *(const v2f*)(aptr + 4 * (s + 1));
                bn = *(const v2f*)(bptr + 4 * (s + 1));
            }
            // D = A*B + C, full f32 (neg_a, A, neg_b, B, c_mod, C, reuse_a, reuse_b)
            acc = __builtin_amdgcn_wmma_f32_16x16x4_f32(
                      false, a, false, bf, (short)0, acc, false, false);
            if (s < 31) { a = an; bf = bn; }
        }

        // D layout: VGPR v, lanes 0-15 -> M=v, lanes 16-31 -> M=8+v; N = lane&15.
        // Columns 0 and 1 hold the two score vectors for this bank.
        if (n < 2) {
            const int slot = (n == 0) ? slot0 : slot1;
            const long long base = (long long)slot * BATCH * KP1 + (long long)b * KP1;
            const int k0 = kt * 16 + chi * 8;
            float* __restrict__ dst = outs + base + k0;
            if (k0 + 7 < KP1) {
                #pragma unroll
                for (int v = 0; v < 8; ++v) dst[v] = acc[v];
            } else {
                #pragma unroll
                for (int v = 0; v < 8; ++v)
                    if (k0 + v < KP1) dst[v] = acc[v];
            }
        }
    }
}

// ---------------------------------------------------------------------------
// float4 copy of a memory bank into the output region.
// ---------------------------------------------------------------------------
__global__ void copy_kernel(const float4* __restrict__ src,
                            float4* __restrict__ dst, long long n4)
{
    long long i = (long long)blockIdx.x * blockDim.x + threadIdx.x;
    long long stride = (long long)gridDim.x * blockDim.x;
    for (; i < n4; i += stride) dst[i] = src[i];
}

// ---------------------------------------------------------------------------
// Momentum update of rows y[b]: pos = 0.5*mem + 0.5*feat, L2-normalized.
// Deterministic "last duplicate wins" like jax .at[y].set().
// ---------------------------------------------------------------------------
__global__ __launch_bounds__(DIM) void update_kernel(
    const float* __restrict__ l, const float* __restrict__ ab,
    const float* __restrict__ ori, const int* __restrict__ y,
    const float* __restrict__ ml, const float* __restrict__ mab,
    const float* __restrict__ mori,
    float* __restrict__ nl, float* __restrict__ nab, float* __restrict__ nori)
{
    const int b = blockIdx.x, bank = blockIdx.y, t = threadIdx.x;
    const float* feat; const float* mem; float* dst;
    if (bank == 0)      { feat = l;   mem = ml;   dst = nl;   }
    else if (bank == 1) { feat = ab;  mem = mab;  dst = nab;  }
    else                { feat = ori; mem = mori; dst = nori; }

    const int row = y[b];
    for (int b2 = b + 1; b2 < BATCH; ++b2)
        if (y[b2] == row) return;   // later batch owns this row (uniform exit)

    float pos = mem[(long long)row * DIM + t] * 0.5f + feat[b * DIM + t] * 0.5f;

    __shared__ float red[DIM];
    red[t] = pos * pos;
    __syncthreads();
    for (int s = DIM / 2; s > 0; s >>= 1) {
        if (t < s) red[t] += red[t + s];
        __syncthreads();
    }
    dst[(long long)row * DIM + t] = pos / sqrtf(red[0]);
}

// ---------------------------------------------------------------------------
extern "C" void kernel_launch(void* const* d_in, const int* in_sizes, int n_in,
                              void* d_out, int out_size, void* d_ws, size_t ws_size,
                              hipStream_t stream)
{
    const float* l       = (const float*)d_in[0];
    const float* ab      = (const float*)d_in[1];
    const float* ori     = (const float*)d_in[2];
    const int*   y       = (const int*)d_in[3];
    const int*   idx     = (const int*)d_in[4];
    const float* mem_l   = (const float*)d_in[5];
    const float* mem_ab  = (const float*)d_in[6];
    const float* mem_ori = (const float*)d_in[7];

    float* out     = (float*)d_out;
    float* outs    = out;                       // (6, B, K+1)
    float* new_l   = out + OUTS_SZ;             // (200000, 128)
    float* new_ab  = new_l + MEMSZ;
    float* new_ori = new_ab + MEMSZ;

    // 1) copy banks into output (momentum rows overwritten afterwards)
    const long long n4 = MEMSZ / 4;
    const int cgrid = (int)((n4 + 255) / 256);
    copy_kernel<<<cgrid, 256, 0, stream>>>((const float4*)mem_l,   (float4*)new_l,   n4);
    copy_kernel<<<cgrid, 256, 0, stream>>>((const float4*)mem_ab,  (float4*)new_ab,  n4);
    copy_kernel<<<cgrid, 256, 0, stream>>>((const float4*)mem_ori, (float4*)new_ori, n4);

    // 2) gathered WMMA scores: grid = (k-tile blocks, batch, bank)
    dim3 sgrid((NTILE + WAVES_PER_BLOCK - 1) / WAVES_PER_BLOCK, BATCH, 3);
    score_kernel<<<sgrid, WAVES_PER_BLOCK * 32, 0, stream>>>(
        l, ab, ori, idx, mem_l, mem_ab, mem_ori, outs);

    // 3) momentum + normalize scatter (after copies on the same stream)
    dim3 ugrid(BATCH, 3);
    update_kernel<<<ugrid, DIM, 0, stream>>>(
        l, ab, ori, y, mem_l, mem_ab, mem_ori, new_l, new_ab, new_ori);
}